// Siamese_GeoCheby_Cos_54451595379149
// MI455X (gfx1250) — compile-verified
//
#include <hip/hip_runtime.h>
#include <hip/hip_bf16.h>

#define NNODE  268
#define NPAD   272        // 17 WMMA M-tiles
#define NEDGE  8576
#define NBATCH 64
#define HID    16
#define FCK    4288       // 268*16
#define FC1N   1024
#define FC2N   512
#define YS     33         // ybuf row stride (odd -> LDS banks spread)
#define AS     17         // accumulator row stride (odd -> LDS banks spread)

typedef __attribute__((ext_vector_type(2))) float v2f;
typedef __attribute__((ext_vector_type(8))) float v8f;

// D = A(16x4 f32) * B(4x16 f32) + C(16x16 f32), full fp32 on the WMMA pipe.
__device__ __forceinline__ v8f wmma4(v2f a, v2f b, v8f c) {
  return __builtin_amdgcn_wmma_f32_16x16x4_f32(false, a, false, b, (short)0, c,
                                               false, false);
}

// ---------------------------------------------------------------------------
// Per-(branch,batch) fused graph pipeline. 8 waves, everything in LDS.
// smem: ybuf[272*33] | accA[272*17] | accB[272*17] | nrm[8576] | erc[8576] | degs[272]
// ---------------------------------------------------------------------------
__global__ __launch_bounds__(256)
void gc_kernel(const float* __restrict__ x1, const float* __restrict__ x2,
               const float* __restrict__ x3,
               const int* __restrict__ ei1, const int* __restrict__ ei2,
               const int* __restrict__ ei3,
               const float* __restrict__ ea1, const float* __restrict__ ea2,
               const float* __restrict__ ea3,
               const float* __restrict__ gc1w, const float* __restrict__ gc1b,
               const float* __restrict__ gc4w, const float* __restrict__ gc4b,
               float* __restrict__ cbuf /* [3][64][4288] */) {
  extern __shared__ float smem[];
  float* ybuf = smem;                    // NPAD*YS
  float* accA = ybuf + NPAD * YS;        // NPAD*AS
  float* accB = accA + NPAD * AS;        // NPAD*AS
  float* nrm  = accB + NPAD * AS;        // NEDGE
  int*   erc  = (int*)(nrm + NEDGE);     // NEDGE packed (r<<16)|c
  float* degs = (float*)(erc + NEDGE);   // NPAD

  const int br = blockIdx.x / NBATCH;
  const int bt = blockIdx.x % NBATCH;
  const float* x  = (br == 0 ? x1 : br == 1 ? x2 : x3) + (size_t)bt * NNODE * NNODE;
  const int*   ei = (br == 0 ? ei1 : br == 1 ? ei2 : ei3) + (size_t)bt * 2 * NEDGE;
  const float* ea = (br == 0 ? ea1 : br == 1 ? ea2 : ea3) + (size_t)bt * NEDGE;
  float* cout = cbuf + ((size_t)br * NBATCH + bt) * FCK;

  const int tid = threadIdx.x;

  // ---- Chebyshev normalization: deg -> rsqrt -> norm (all in LDS) ----
  for (int i = tid; i < NPAD; i += 256) degs[i] = 0.f;
  __syncthreads();
  for (int e = tid; e < NEDGE; e += 256) atomicAdd(&degs[ei[e]], ea[e]);
  __syncthreads();
  for (int i = tid; i < NPAD; i += 256) {
    float d = degs[i];
    degs[i] = (d > 0.f) ? rsqrtf(d) : 0.f;
  }
  __syncthreads();
  for (int e = tid; e < NEDGE; e += 256) {
    int r = ei[e], c = ei[NEDGE + e];
    nrm[e] = -ea[e] * degs[r] * degs[c];
    erc[e] = (r << 16) | c;              // LDS-resident edge list for scatters
  }
  for (int i = tid; i < NPAD * AS; i += 256) accA[i] = 0.f;
  __syncthreads();

  const int wave = tid >> 5, lane = tid & 31;
  const int mloc = lane & 15;
  const int kh   = (lane >> 4) << 1;   // f32 WMMA A/B: lanes 16-31 hold K+2,K+3

  // ---- y[272][32] = x @ [W0 | W1]   (M=272pad, N=32, K=268) ----
  // Full tiles (rows 0..255): maskless. Last M-tile handled by waves 0-1 with
  // clamped address + 0/1 mask (wave-uniform branch, EXEC stays all-ones).
  for (int t = wave; t < 32; t += 8) {
    int mt = t >> 1, nt = t & 1;
    const float* xr = x + (size_t)(mt * 16 + mloc) * NNODE + kh;
    const float* wp = gc1w + nt * (NNODE * HID) + kh * HID + mloc;
    v8f acc = {};
    for (int kk = 0; kk < NNODE; kk += 4) {
      v2f a = *(const v2f*)(xr + kk);
      v2f b;
      b.x = wp[kk * HID];
      b.y = wp[kk * HID + HID];
      acc = wmma4(a, b, acc);
    }
    int rbase = mt * 16 + ((lane >> 4) << 3);
#pragma unroll
    for (int j = 0; j < 8; j++)
      ybuf[(rbase + j) * YS + nt * 16 + mloc] = acc[j];
  }
  if (wave < 2) {                        // mt = 16, nt = wave
    int nt = wave;
    int row = 256 + mloc;
    const float msk = (row < NNODE) ? 1.f : 0.f;
    const int rowc = (row < NNODE) ? row : (NNODE - 1);
    const float* xr = x + (size_t)rowc * NNODE + kh;
    const float* wp = gc1w + nt * (NNODE * HID) + kh * HID + mloc;
    v8f acc = {};
    for (int kk = 0; kk < NNODE; kk += 4) {
      v2f a = *(const v2f*)(xr + kk);
      a.x *= msk;
      a.y *= msk;
      v2f b;
      b.x = wp[kk * HID];
      b.y = wp[kk * HID + HID];
      acc = wmma4(a, b, acc);
    }
    int rbase = 256 + ((lane >> 4) << 3);
#pragma unroll
    for (int j = 0; j < 8; j++)
      ybuf[(rbase + j) * YS + nt * 16 + mloc] = acc[j];
  }
  __syncthreads();

  // ---- accA[r][:] += nrm[e] * y1[c][:]  (conflict-spread LDS scatter) ----
  for (int e = tid; e < NEDGE; e += 256) {
    int u = erc[e];
    int r = u >> 16, c = u & 0xffff;
    float w = nrm[e];
    const float* yr = ybuf + c * YS + 16;
    float* ar = accA + r * AS;
#pragma unroll
    for (int j = 0; j < 16; j++) atomicAdd(&ar[j], w * yr[j]);
  }
  __syncthreads();

  // ---- h = relu(y0 + accA + b1) -> accA (zero pad rows); accB = 0 ----
  for (int i = tid; i < NPAD * HID; i += 256) {
    int n = i >> 4, j = i & 15;
    float v = 0.f;
    if (n < NNODE) {
      v = ybuf[n * YS + j] + accA[n * AS + j] + gc1b[j];
      v = v > 0.f ? v : 0.f;
    }
    accA[n * AS + j] = v;
    accB[n * AS + j] = 0.f;
  }
  __syncthreads();

  // ---- z[272][32] = h @ [W0' | W1']   (K=16, h zero-padded, LDS A reads
  //      conflict-free: 17*row mod 64 distinct for row 0..15) ----
  for (int t = wave; t < 34; t += 8) {
    int mt = t >> 1, nt = t & 1;
    const float* hr = accA + (mt * 16 + mloc) * AS + kh;
    const float* wp = gc4w + nt * (HID * HID) + kh * HID + mloc;
    v8f acc = {};
#pragma unroll
    for (int kk = 0; kk < HID; kk += 4) {
      v2f a, b;
      a.x = hr[kk];
      a.y = hr[kk + 1];
      b.x = wp[kk * HID];
      b.y = wp[kk * HID + HID];
      acc = wmma4(a, b, acc);
    }
    int rbase = mt * 16 + ((lane >> 4) << 3);
#pragma unroll
    for (int j = 0; j < 8; j++)
      ybuf[(rbase + j) * YS + nt * 16 + mloc] = acc[j];
  }
  __syncthreads();

  // ---- accB[r][:] += nrm[e] * z1[c][:] ----
  for (int e = tid; e < NEDGE; e += 256) {
    int u = erc[e];
    int r = u >> 16, c = u & 0xffff;
    float w = nrm[e];
    const float* zr = ybuf + c * YS + 16;
    float* ar = accB + r * AS;
#pragma unroll
    for (int j = 0; j < 16; j++) atomicAdd(&ar[j], w * zr[j]);
  }
  __syncthreads();

  // ---- c = z0 + accB + b4 (no relu), flattened [268*16] ----
  for (int i = tid; i < NNODE * HID; i += 256) {
    int n = i >> 4, j = i & 15;
    cout[i] = ybuf[n * YS + j] + accB[n * AS + j] + gc4b[j];
  }
}

// ---------------------------------------------------------------------------
// FC1: per branch, [64,4288] @ [4288,1024]^T + b, relu.  48 blocks x 8 waves.
// K-loop unrolled 4x: clauses of global_load_b64 feeding pipelined WMMAs.
// ---------------------------------------------------------------------------
__global__ __launch_bounds__(256)
void fc1_kernel(const float* __restrict__ cbuf, const float* __restrict__ w,
                const float* __restrict__ bias, float* __restrict__ a1) {
  const int br = blockIdx.x >> 4;
  const int chunk = blockIdx.x & 15;
  const int wave = threadIdx.x >> 5, lane = threadIdx.x & 31;
  const int mloc = lane & 15, kh = (lane >> 4) << 1;
  const float* A = cbuf + (size_t)br * NBATCH * FCK;
  float* out = a1 + (size_t)br * NBATCH * FC1N;

  for (int tt = wave; tt < 16; tt += 8) {
    int mt = tt >> 2;
    int nt = chunk * 4 + (tt & 3);
    int m = mt * 16 + mloc;            // < 64 always
    int n = nt * 16 + mloc;
    const float* ap = A + (size_t)m * FCK + kh;
    const float* bp = w + (size_t)n * FCK + kh;
    v8f acc = {};
#pragma unroll 4
    for (int kk = 0; kk < FCK; kk += 4) {
      v2f a = *(const v2f*)(ap + kk);
      v2f b = *(const v2f*)(bp + kk);
      acc = wmma4(a, b, acc);
    }
    int rbase = mt * 16 + ((lane >> 4) << 3);
    float bn = bias[n];
#pragma unroll
    for (int j = 0; j < 8; j++) {
      float v = acc[j] + bn;
      out[(size_t)(rbase + j) * FC1N + n] = v > 0.f ? v : 0.f;
    }
  }
}

// ---------------------------------------------------------------------------
// FC2: per branch, [64,1024] @ [1024,512]^T + b -> d_out.  48 blocks x 8 waves.
// ---------------------------------------------------------------------------
__global__ __launch_bounds__(256)
void fc2_kernel(const float* __restrict__ a1, const float* __restrict__ w,
                const float* __restrict__ bias, float* __restrict__ out) {
  const int br = blockIdx.x >> 4;
  const int chunk = blockIdx.x & 15;
  const int wave = threadIdx.x >> 5, lane = threadIdx.x & 31;
  const int mloc = lane & 15, kh = (lane >> 4) << 1;
  const float* A = a1 + (size_t)br * NBATCH * FC1N;
  float* O = out + (size_t)br * NBATCH * FC2N;

  int t = chunk * 8 + wave;            // 0..127 tiles per branch
  int mt = t >> 5, nt = t & 31;
  int m = mt * 16 + mloc;
  int n = nt * 16 + mloc;
  const float* ap = A + (size_t)m * FC1N + kh;
  const float* bp = w + (size_t)n * FC1N + kh;
  v8f acc = {};
#pragma unroll 4
  for (int kk = 0; kk < FC1N; kk += 4) {
    v2f a = *(const v2f*)(ap + kk);
    v2f b = *(const v2f*)(bp + kk);
    acc = wmma4(a, b, acc);
  }
  int rbase = mt * 16 + ((lane >> 4) << 3);
  float bn = bias[n];
#pragma unroll
  for (int j = 0; j < 8; j++)
    O[(size_t)(rbase + j) * FC2N + n] = acc[j] + bn;
}

extern "C" void kernel_launch(void* const* d_in, const int* in_sizes, int n_in,
                              void* d_out, int out_size, void* d_ws, size_t ws_size,
                              hipStream_t stream) {
  const float* x1  = (const float*)d_in[0];
  const float* x2  = (const float*)d_in[1];
  const float* x3  = (const float*)d_in[2];
  const int*   ei1 = (const int*)d_in[3];
  const int*   ei2 = (const int*)d_in[4];
  const int*   ei3 = (const int*)d_in[5];
  const float* ea1 = (const float*)d_in[6];
  const float* ea2 = (const float*)d_in[7];
  const float* ea3 = (const float*)d_in[8];
  const float* gc1w = (const float*)d_in[9];
  const float* gc1b = (const float*)d_in[10];
  const float* gc4w = (const float*)d_in[11];
  const float* gc4b = (const float*)d_in[12];
  const float* fc1w = (const float*)d_in[13];
  const float* fc1b = (const float*)d_in[14];
  const float* fc2w = (const float*)d_in[15];
  const float* fc2b = (const float*)d_in[16];

  float* ws   = (float*)d_ws;
  float* cbuf = ws;                                  // 3*64*4288 floats
  float* a1   = ws + (size_t)3 * NBATCH * FCK;       // 3*64*1024 floats

  size_t smem = (size_t)(NPAD * YS + 2 * NPAD * AS + 2 * NEDGE + NPAD) * sizeof(float);
  gc_kernel<<<dim3(3 * NBATCH), dim3(256), smem, stream>>>(
      x1, x2, x3, ei1, ei2, ei3, ea1, ea2, ea3,
      gc1w, gc1b, gc4w, gc4b, cbuf);
  fc1_kernel<<<dim3(48), dim3(256), 0, stream>>>(cbuf, fc1w, fc1b, a1);
  fc2_kernel<<<dim3(48), dim3(256), 0, stream>>>(a1, fc2w, fc2b, (float*)d_out);
}